// CharLevelModel_83966610637044
// MI455X (gfx1250) — compile-verified
//
#include <hip/hip_runtime.h>

#define B_   64
#define T_   256
#define D_   768
#define C_   141
#define H1_  282
#define TWOD 1536

typedef __attribute__((ext_vector_type(2))) float v2f;
typedef __attribute__((ext_vector_type(8))) float v8f;

// ---------------------------------------------------------------------------
// K1: char -> token map. One block per batch b; offsets staged in LDS; each
// char c scans all T spans, last (max) hit wins, -1 if none.
// ---------------------------------------------------------------------------
__global__ __launch_bounds__(256) void tok_kernel(const int* __restrict__ om,
                                                  int* __restrict__ tokbuf) {
  __shared__ int lo_s[T_];
  __shared__ int hi_s[T_];
  const int b = blockIdx.x;
  const int t = threadIdx.x;
  lo_s[t] = om[(b * T_ + t) * 2 + 0];
  hi_s[t] = om[(b * T_ + t) * 2 + 1];
  __syncthreads();
  const int c = threadIdx.x;
  if (c < C_) {
    int tok = -1;
#pragma unroll 4
    for (int tt = 0; tt < T_; ++tt) {
      if (lo_s[tt] <= c && c < hi_s[tt]) tok = tt;  // ascending scan => max t
    }
    tokbuf[b * C_ + c] = tok;
  }
}

// ---------------------------------------------------------------------------
// K2: G = W1 @ W2   (1536 x 2). Tiny (0.9 MFLOP): one thread per output.
// ---------------------------------------------------------------------------
__global__ __launch_bounds__(256) void g_kernel(const float* __restrict__ W1,
                                                const float* __restrict__ W2,
                                                float* __restrict__ G) {
  const int idx = blockIdx.x * 256 + threadIdx.x;  // 0..3071
  if (idx >= TWOD * 2) return;
  const int m = idx >> 1;
  const int j = idx & 1;
  float acc = 0.f;
  for (int h = 0; h < H1_; ++h) acc += W1[m * H1_ + h] * W2[h * 2 + j];
  G[idx] = acc;
}

// ---------------------------------------------------------------------------
// K3: Wfold = W_tok @ G via V_WMMA_F32_16X16X4_F32, N padded 2 -> 16.
// One wave per 16-row M tile; K loop over 1536 in steps of 4 (384 WMMAs).
// A layout: lane&15 = M row, lane>=16 selects K pair {2,3} vs {0,1}.
// B layout: VGPR v holds row K=v (lanes 0-15) / K=v+2 (lanes 16-31), N=lane&15.
// D layout: acc[r] holds M = m0 + r + 8*(lane>=16), N = lane&15.
//
// Padding columns 2..15 are produced with UNCONDITIONAL loads (lanes n>=2
// alias lane n&1's address, which is always valid and L2-hot) followed by a
// v_cndmask select against a loop-invariant predicate — this avoids the
// exec-mask save/restore sequences the previous version generated.
// ---------------------------------------------------------------------------
__global__ __launch_bounds__(32) void wfold_wmma_kernel(const float* __restrict__ Wtok,
                                                        const float* __restrict__ G,
                                                        float* __restrict__ Wfold) {
  const int m0   = blockIdx.x * 16;      // 96 tiles
  const int lane = threadIdx.x;          // wave32
  const int mloc = lane & 15;
  const int h    = lane >> 4;            // K-half select (0 or 1)
  const int n    = lane & 15;            // N column
  const bool live = (n < 2);
  const int  col  = n & 1;

  const float* arow = Wtok + (size_t)(m0 + mloc) * TWOD + 2 * h;
  // gcol[2k]   = G[(k + 2h    ) * 2 + col]
  // gcol[2k+2] = G[(k + 2h + 1) * 2 + col]
  const float* gcol = G + 4 * h + col;

  v8f acc = {};
#pragma unroll 8
  for (int k = 0; k < TWOD; k += 4) {
    v2f a;
    a.x = arow[k];
    a.y = arow[k + 1];
    const float g0 = gcol[2 * k];
    const float g1 = gcol[2 * k + 2];
    v2f bm;
    bm.x = live ? g0 : 0.0f;
    bm.y = live ? g1 : 0.0f;
    acc = __builtin_amdgcn_wmma_f32_16x16x4_f32(
        /*neg_a=*/false, a, /*neg_b=*/false, bm,
        /*c_mod=*/(short)0, acc, /*reuse_a=*/false, /*reuse_b=*/false);
  }

  if (live) {
#pragma unroll
    for (int r = 0; r < 8; ++r) {
      const int m = m0 + r + 8 * h;
      Wfold[m * 2 + col] = acc[r];
    }
  }
}

// ---------------------------------------------------------------------------
// K4: folded biases. biases[0..1] = b_tok@G + b1@W2 + b2 (valid path),
//     biases[2..3] = b1@W2 + b2 (invalid path).
// ---------------------------------------------------------------------------
__global__ __launch_bounds__(32) void bias_kernel(const float* __restrict__ b_tok,
                                                  const float* __restrict__ b1,
                                                  const float* __restrict__ W2,
                                                  const float* __restrict__ b2,
                                                  const float* __restrict__ G,
                                                  float* __restrict__ biases) {
  const int j = threadIdx.x;
  if (j >= 2) return;
  float bi = b2[j];
  for (int h = 0; h < H1_; ++h) bi += b1[h] * W2[h * 2 + j];
  float bv = bi;
  for (int k = 0; k < TWOD; ++k) bv += b_tok[k] * G[k * 2 + j];
  biases[j]     = bv;
  biases[2 + j] = bi;
}

// ---------------------------------------------------------------------------
// K5: one wave32 per (b,c): gather row H[b,tok] from h0/h1 (float4 coalesced)
// and dot with the two Wfold columns; shuffle-reduce; lane 0 writes.
// ---------------------------------------------------------------------------
__global__ __launch_bounds__(256) void out_kernel(const float* __restrict__ h0,
                                                  const float* __restrict__ h1,
                                                  const int* __restrict__ tokbuf,
                                                  const float* __restrict__ Wfold,
                                                  const float* __restrict__ biases,
                                                  float* __restrict__ out) {
  const int wid  = blockIdx.x * 8 + (threadIdx.x >> 5);
  const int lane = threadIdx.x & 31;
  if (wid >= B_ * C_) return;

  __builtin_prefetch(Wfold, 0, 3);  // keep the 12 KB folded weight hot

  const int b   = wid / C_;
  const int tok = tokbuf[wid];
  if (tok < 0) {
    if (lane == 0) {
      out[wid]           = biases[2];
      out[B_ * C_ + wid] = biases[3];
    }
    return;
  }

  const float* r0 = h0 + (size_t)(b * T_ + tok) * D_;
  const float* r1 = h1 + (size_t)(b * T_ + tok) * D_;

  float acc0 = 0.f, acc1 = 0.f;
  for (int base = lane * 4; base < D_; base += 128) {
    const float4 x0 = *(const float4*)(r0 + base);
    const float4 x1 = *(const float4*)(r1 + base);
    const float4 wa = *(const float4*)(Wfold + base * 2);            // rows base, base+1
    const float4 wb = *(const float4*)(Wfold + base * 2 + 4);        // rows base+2, base+3
    const float4 wc = *(const float4*)(Wfold + (D_ + base) * 2);     // h1 half
    const float4 wd = *(const float4*)(Wfold + (D_ + base) * 2 + 4);
    acc0 += x0.x * wa.x + x0.y * wa.z + x0.z * wb.x + x0.w * wb.z;
    acc1 += x0.x * wa.y + x0.y * wa.w + x0.z * wb.y + x0.w * wb.w;
    acc0 += x1.x * wc.x + x1.y * wc.z + x1.z * wd.x + x1.w * wd.z;
    acc1 += x1.x * wc.y + x1.y * wc.w + x1.z * wd.y + x1.w * wd.w;
  }

#pragma unroll
  for (int off = 16; off > 0; off >>= 1) {
    acc0 += __shfl_down(acc0, off, 32);
    acc1 += __shfl_down(acc1, off, 32);
  }

  if (lane == 0) {
    out[wid]           = acc0 + biases[0];
    out[B_ * C_ + wid] = acc1 + biases[1];
  }
}

// ---------------------------------------------------------------------------
extern "C" void kernel_launch(void* const* d_in, const int* in_sizes, int n_in,
                              void* d_out, int out_size, void* d_ws, size_t ws_size,
                              hipStream_t stream) {
  const float* hidden0 = (const float*)d_in[0];
  const float* hidden1 = (const float*)d_in[1];
  const int*   om      = (const int*)d_in[2];
  const float* W_tok   = (const float*)d_in[3];
  const float* b_tok   = (const float*)d_in[4];
  const float* W1      = (const float*)d_in[5];
  const float* b1      = (const float*)d_in[6];
  const float* W2      = (const float*)d_in[7];
  const float* b2      = (const float*)d_in[8];
  const float* hstate  = (const float*)d_in[9];
  float* out = (float*)d_out;

  char* ws = (char*)d_ws;
  int*   tokbuf = (int*)(ws);                    // B*C ints   (36 KB)
  float* G      = (float*)(ws + 40960);          // 1536x2     (12 KB)
  float* Wfold  = (float*)(ws + 40960 + 12288);  // 1536x2     (12 KB)
  float* biases = (float*)(ws + 40960 + 24576);  // 4 floats

  // hidden_state passthrough: outputs [out0 | out1 | hidden_state]
  hipMemcpyAsync(out + 2 * B_ * C_, hstate, sizeof(float) * 16 * B_ * D_,
                 hipMemcpyDeviceToDevice, stream);

  tok_kernel<<<B_, 256, 0, stream>>>(om, tokbuf);
  g_kernel<<<(TWOD * 2 + 255) / 256, 256, 0, stream>>>(W1, W2, G);
  wfold_wmma_kernel<<<TWOD / 16, 32, 0, stream>>>(W_tok, G, Wfold);
  bias_kernel<<<1, 32, 0, stream>>>(b_tok, b1, W2, b2, G, biases);
  out_kernel<<<(B_ * C_ + 7) / 8, 256, 0, stream>>>(hidden0, hidden1, tokbuf,
                                                    Wfold, biases, out);
}